// ModifiedSpecformer_74259984548332
// MI455X (gfx1250) — compile-verified
//
#include <hip/hip_runtime.h>
#include <math.h>

// Problem constants (from reference)
#define BB   2
#define NN   512
#define DD   128
#define KK   8
#define HH   8
#define HD   16
#define T2N  1024   // 2*N

typedef __attribute__((ext_vector_type(2))) float v2f;
typedef __attribute__((ext_vector_type(8))) float v8f;

// ---- CDNA5 fp32 WMMA: D(16x16) = A(16x4) * B(4x16) + C --------------------
__device__ __forceinline__ v8f wmma4(v2f a, v2f b, v8f c) {
  // (neg_a, A, neg_b, B, c_mod, C, reuse_a, reuse_b)
  return __builtin_amdgcn_wmma_f32_16x16x4_f32(false, a, false, b, (short)0, c,
                                               false, false);
}

__device__ __forceinline__ float gelu_f(float x) {
  return 0.5f * x * (1.0f + erff(x * 0.70710678118654752f));
}

// ---------------------------------------------------------------------------
// Generic WMMA GEMM: C[M x Nc] = epi( preA(A)[M x Kd] @ B[Kd x Nc] )
// One wave (32 threads) per block computes a 64x16 tile of C (4 row-subtiles
// sharing each B fragment). Fragment layouts (wave32):
//   A frag : lane l -> row = l&15, k = (l>>4)*2 + v   (v = vgpr 0/1)
//   B frag : lane l -> col = l&15, k = (l>>4)*2 + v
//   C frag : vgpr r -> row = r + (l>>4)*8, col = l&15
// PREA: 0 none | 1 gelu(a*x+c) | 2 relu(a*x+c)   (BN folded to affine)
// TRANSB: B indexed as Bm[col*ldb + k]  (i.e. logical B = Bm^T)
// ---------------------------------------------------------------------------
template <int PREA, int TRANSB>
__global__ void gemm16(const float* __restrict__ A, int lda,
                       const float* __restrict__ Bm, int ldb,
                       float* __restrict__ C, int ldc,
                       int M, int Nc, int Kd,
                       const float* __restrict__ bias,
                       const float* __restrict__ resid, int ldres,
                       const float* __restrict__ preScale,
                       const float* __restrict__ preShift,
                       const float* __restrict__ aColScale,
                       int outAct) {
  const int lane  = threadIdx.x & 31;
  const int half  = lane >> 4;
  const int l16   = lane & 15;
  const int row0  = blockIdx.x * 64;
  const int col0  = blockIdx.y * 16;

  v8f acc[4];
  acc[0] = {}; acc[1] = {}; acc[2] = {}; acc[3] = {};

  for (int k = 0; k < Kd; k += 4) {
    const int ka = k + half * 2;
    v2f bf;
    if (TRANSB) {
      const float* bp = Bm + (size_t)(col0 + l16) * ldb + ka;
      bf.x = bp[0]; bf.y = bp[1];
    } else {
      bf.x = Bm[(size_t)(ka + 0) * ldb + col0 + l16];
      bf.y = Bm[(size_t)(ka + 1) * ldb + col0 + l16];
    }
#pragma unroll
    for (int t = 0; t < 4; ++t) {
      const float* arow = A + (size_t)(row0 + t * 16 + l16) * lda;
      if (t == 0 && k + 8 < Kd) __builtin_prefetch(arow + k + 8, 0, 0);
      v2f af; af.x = arow[ka]; af.y = arow[ka + 1];
      if (aColScale) { af.x *= aColScale[ka]; af.y *= aColScale[ka + 1]; }
      if (PREA == 1) {
        af.x = gelu_f(af.x * preScale[ka] + preShift[ka]);
        af.y = gelu_f(af.y * preScale[ka + 1] + preShift[ka + 1]);
      } else if (PREA == 2) {
        af.x = fmaxf(af.x * preScale[ka] + preShift[ka], 0.0f);
        af.y = fmaxf(af.y * preScale[ka + 1] + preShift[ka + 1], 0.0f);
      }
      acc[t] = wmma4(af, bf, acc[t]);
    }
  }

  const int n = col0 + l16;
  const float bn = bias ? bias[n] : 0.0f;
#pragma unroll
  for (int t = 0; t < 4; ++t) {
#pragma unroll
    for (int r = 0; r < 8; ++r) {
      const int m = row0 + t * 16 + r + half * 8;
      float vv = acc[t][r] + bn;
      if (resid) vv += resid[(size_t)m * ldres + n];
      if (outAct == 1) vv = gelu_f(vv);
      else if (outAct == 2) vv = fmaxf(vv, 0.0f);
      C[(size_t)m * ldc + n] = vv;
    }
  }
}

// ---------------------------------------------------------------------------
// Flash attention: one wave per (b, h, 16-query tile). hd = 16.
// q/k/v: (B, 2N, D) row-major, head h occupies columns [h*16, h*16+16).
// ---------------------------------------------------------------------------
__global__ void attn16(const float* __restrict__ q, const float* __restrict__ k,
                       const float* __restrict__ v,
                       const int* __restrict__ length,
                       float* __restrict__ o) {
  __shared__ float pt[16][17];
  const int lane = threadIdx.x & 31;
  const int half = lane >> 4;
  const int l16  = lane & 15;
  const int b    = blockIdx.y / HH;
  const int h    = blockIdx.y % HH;
  const int q0   = blockIdx.x * 16;
  const int len  = length[b];

  const float* qb = q + ((size_t)b * T2N) * DD + h * HD;
  const float* kb = k + ((size_t)b * T2N) * DD + h * HD;
  const float* vb = v + ((size_t)b * T2N) * DD + h * HD;

  // Preload Q A-fragments for the 4 k-steps of hd=16
  v2f qa[4];
#pragma unroll
  for (int ks = 0; ks < 4; ++ks) {
    const int ka = ks * 4 + half * 2;
    const float* row = qb + (size_t)(q0 + l16) * DD;
    qa[ks].x = row[ka]; qa[ks].y = row[ka + 1];
  }

  v8f oacc = {};
  float mr[8], lr[8];
#pragma unroll
  for (int r = 0; r < 8; ++r) { mr[r] = -1e30f; lr[r] = 0.0f; }

  for (int kt = 0; kt < T2N / 16; ++kt) {
    // S = Q @ K^T   (B frag: col = key = l16, k = feature)
    v8f s = {};
#pragma unroll
    for (int ks = 0; ks < 4; ++ks) {
      const int ka = ks * 4 + half * 2;
      const float* krow = kb + (size_t)(kt * 16 + l16) * DD;
      v2f bf; bf.x = krow[ka]; bf.y = krow[ka + 1];
      s = wmma4(qa[ks], bf, s);
    }
    const int kg = kt * 16 + l16;  // this lane's key column
    const float madd = (kg < NN && kg >= len) ? -1.0e9f : 0.0f;

    __syncthreads();
#pragma unroll
    for (int r = 0; r < 8; ++r) {
      float sv = s[r] * 0.25f + madd;  // scale = hd^-0.5
      // row max / sum over the 16 lanes holding this row
      float mx = sv;
#pragma unroll
      for (int msk = 1; msk < 16; msk <<= 1) mx = fmaxf(mx, __shfl_xor(mx, msk, 32));
      const float mnew = fmaxf(mr[r], mx);
      const float p = __expf(sv - mnew);
      float ps = p;
#pragma unroll
      for (int msk = 1; msk < 16; msk <<= 1) ps += __shfl_xor(ps, msk, 32);
      const float corr = __expf(mr[r] - mnew);
      lr[r] = lr[r] * corr + ps;
      mr[r] = mnew;
      oacc[r] *= corr;
      pt[r + half * 8][l16] = p;  // C-layout -> LDS (row = query, col = key)
    }
    __syncthreads();

    // O += P @ V  (A frag from LDS: row = query = l16, k = key)
#pragma unroll
    for (int ks = 0; ks < 4; ++ks) {
      const int ka = ks * 4 + half * 2;
      v2f af; af.x = pt[l16][ka]; af.y = pt[l16][ka + 1];
      v2f bf;
      bf.x = vb[(size_t)(kt * 16 + ka + 0) * DD + l16];
      bf.y = vb[(size_t)(kt * 16 + ka + 1) * DD + l16];
      oacc = wmma4(af, bf, oacc);
    }
  }

  float* ob = o + ((size_t)b * T2N + q0) * DD + h * HD;
#pragma unroll
  for (int r = 0; r < 8; ++r)
    ob[(size_t)(r + half * 8) * DD + l16] = oacc[r] / lr[r];
}

// ---------------------------------------------------------------------------
// Small helpers
// ---------------------------------------------------------------------------
__global__ void concat_seq(const float* __restrict__ eig,
                           const float* __restrict__ nf,
                           float* __restrict__ seq) {
  const int idx = blockIdx.x * blockDim.x + threadIdx.x;  // B*N*D
  const int d = idx % DD;
  const int r = idx / DD;
  const int b = r / NN, n = r % NN;
  seq[((size_t)b * T2N + n) * DD + d]      = eig[idx];
  seq[((size_t)b * T2N + NN + n) * DD + d] = nf[idx];
}

// new_e[b,k,n] = sum_d eig2[b,n,d] * Wd[d,k] + bd[k]   (eig2 = seq2 first half)
__global__ void newe_k(const float* __restrict__ seq2,
                       const float* __restrict__ Wd,
                       const float* __restrict__ bd,
                       float* __restrict__ ne) {
  const int idx = blockIdx.x * blockDim.x + threadIdx.x;  // B*N*K
  if (idx >= BB * NN * KK) return;
  const int kk = idx % KK;
  const int n  = (idx / KK) % NN;
  const int b  = idx / (KK * NN);
  const float* row = seq2 + ((size_t)b * T2N + n) * DD;
  float s = bd[kk];
#pragma unroll 4
  for (int d = 0; d < DD; ++d) s += row[d] * Wd[d * KK + kk];
  ne[((size_t)b * KK + kk) * NN + n] = s;
}

// h1[r,d] = b1[d] + (i==j)*W1[0,d] + sum_k filt[b,k,i,j]*W1[1+k,d]
__global__ void h1_k(const float* __restrict__ filt,
                     const float* __restrict__ W1,
                     const float* __restrict__ b1,
                     float* __restrict__ h1) {
  __shared__ float W1s[KK + 1][DD];
  __shared__ float b1s[DD];
  const int t = threadIdx.x;  // 128
#pragma unroll
  for (int r = 0; r < KK + 1; ++r) W1s[r][t] = W1[r * DD + t];
  b1s[t] = b1[t];
  __syncthreads();
  const size_t row = (size_t)blockIdx.x * 4;
  for (int rr = 0; rr < 4; ++rr) {
    const size_t r = row + rr;
    const int j = (int)(r % NN);
    const int i = (int)((r / NN) % NN);
    const int b = (int)(r / ((size_t)NN * NN));
    float acc = b1s[t] + (i == j ? W1s[0][t] : 0.0f);
#pragma unroll
    for (int kk = 0; kk < KK; ++kk) {
      const float f = filt[(((size_t)b * KK + kk) * NN + i) * NN + j];
      acc += f * W1s[1 + kk][t];
    }
    h1[r * DD + t] = acc;
  }
}

// Column sum / sumsq of X (M x D) via partials + atomics (buffers pre-zeroed)
__global__ void colstats(const float* __restrict__ X, int M, int chunk,
                         float* __restrict__ sum, float* __restrict__ sq) {
  const int col = threadIdx.x & (DD - 1);
  const int sub = threadIdx.x >> 7;  // blockDim = 256 -> 2 threads per column
  const long r0 = (long)blockIdx.x * chunk;
  float s = 0.0f, q = 0.0f;
  for (int i = sub; i < chunk && r0 + i < M; i += 2) {
    const float x = X[(size_t)(r0 + i) * DD + col];
    s += x; q += x * x;
  }
  atomicAdd(&sum[col], s);
  atomicAdd(&sq[col], q);
}

// Fold BN into per-column affine: bn(x) = a*x + c
__global__ void bn_fin(const float* __restrict__ sum, const float* __restrict__ sq,
                       const float* __restrict__ g, const float* __restrict__ be,
                       float Minv, float* __restrict__ a, float* __restrict__ c) {
  const int d = threadIdx.x;
  const float mean = sum[d] * Minv;
  const float var  = sq[d] * Minv - mean * mean;
  const float rs   = rsqrtf(var + 1e-5f);
  a[d] = g[d] * rs;
  c[d] = be[d] - mean * g[d] * rs;
}

// Fused softmax(axis=j) of gelu(bn2(h2)) with y = sum_j bases*pre ; nf3 = nf2+y
__global__ void smax_y(const float* __restrict__ h2,
                       const float* __restrict__ a2, const float* __restrict__ c2,
                       const float* __restrict__ pre,
                       const float* __restrict__ seq2,
                       float* __restrict__ nf3) {
  const int d = threadIdx.x;           // 128
  const int b = blockIdx.x / NN;
  const int i = blockIdx.x % NN;
  const float* base = h2 + ((size_t)(b * NN + i)) * NN * DD + d;
  const float sa = a2[d], sc = c2[d];
  float mx = -1e30f;
  for (int j = 0; j < NN; ++j)
    mx = fmaxf(mx, gelu_f(base[(size_t)j * DD] * sa + sc));
  float num = 0.0f, den = 0.0f;
  const float* pb = pre + ((size_t)b * NN) * DD + d;
  for (int j = 0; j < NN; ++j) {
    const float s = gelu_f(base[(size_t)j * DD] * sa + sc);
    const float e = __expf(s - mx);
    den += e;
    num += e * pb[(size_t)j * DD];
  }
  const float nf2v = seq2[((size_t)b * T2N + NN + i) * DD + d];
  nf3[((size_t)b * NN + i) * DD + d] = nf2v + num / den;
}

// nf_out = nf3 + relu(bn(t2)) ; eig2 = seq2 first half
__global__ void fin_out(const float* __restrict__ nf3, const float* __restrict__ t2,
                        const float* __restrict__ a4, const float* __restrict__ c4,
                        const float* __restrict__ seq2, float* __restrict__ out) {
  const int idx = blockIdx.x * blockDim.x + threadIdx.x;  // B*N*D
  const int d = idx % DD;
  const int r = idx / DD;
  const int b = r / NN, i = r % NN;
  const float z = fmaxf(t2[idx] * a4[d] + c4[d], 0.0f);
  out[idx] = nf3[idx] + z;
  out[BB * NN * DD + idx] = seq2[((size_t)b * T2N + i) * DD + d];
}

// ---------------------------------------------------------------------------
extern "C" void kernel_launch(void* const* d_in, const int* in_sizes, int n_in,
                              void* d_out, int out_size, void* d_ws, size_t ws_size,
                              hipStream_t stream) {
  (void)in_sizes; (void)n_in; (void)out_size; (void)ws_size;
  const float* eig = (const float*)d_in[0];
  /* d_in[1] == 'e' is unused by the reference forward */
  const float* u   = (const float*)d_in[2];
  const float* nf  = (const float*)d_in[3];
  const int* length = (const int*)d_in[4];
  const float *Wq = (const float*)d_in[5],  *bq = (const float*)d_in[6];
  const float *Wk = (const float*)d_in[7],  *bk = (const float*)d_in[8];
  const float *Wv = (const float*)d_in[9],  *bv = (const float*)d_in[10];
  const float *Wo = (const float*)d_in[11], *bo = (const float*)d_in[12];
  const float *Wd = (const float*)d_in[13], *bd = (const float*)d_in[14];
  const float *W1 = (const float*)d_in[15], *b1 = (const float*)d_in[16];
  const float *g1 = (const float*)d_in[17], *be1 = (const float*)d_in[18];
  const float *W2 = (const float*)d_in[19], *b2 = (const float*)d_in[20];
  const float *g2 = (const float*)d_in[21], *be2 = (const float*)d_in[22];
  const float *Wp = (const float*)d_in[23], *bp = (const float*)d_in[24];
  const float *Wc1 = (const float*)d_in[25], *bc1 = (const float*)d_in[26];
  const float *gc1 = (const float*)d_in[27], *bec1 = (const float*)d_in[28];
  const float *Wc2 = (const float*)d_in[29], *bc2 = (const float*)d_in[30];
  const float *gc2 = (const float*)d_in[31], *bec2 = (const float*)d_in[32];
  float* out = (float*)d_out;

  // ---- workspace layout (floats) ----
  float* w = (float*)d_ws;
  size_t off = 0;
  float* seq  = w + off; off += (size_t)BB * T2N * DD;     // 262144
  float* qb   = w + off; off += (size_t)BB * T2N * DD;
  float* kb   = w + off; off += (size_t)BB * T2N * DD;
  float* vb   = w + off; off += (size_t)BB * T2N * DD;
  float* ob   = w + off; off += (size_t)BB * T2N * DD;
  float* seq2 = w + off; off += (size_t)BB * T2N * DD;
  float* ne   = w + off; off += (size_t)BB * KK * NN;      // 8192
  float* filt = w + off; off += (size_t)BB * KK * NN * NN; // 4.19M
  float* pre  = w + off; off += (size_t)BB * NN * DD;
  float* nf3  = w + off; off += (size_t)BB * NN * DD;
  float* t1   = w + off; off += (size_t)BB * NN * DD;
  float* t2   = w + off; off += (size_t)BB * NN * DD;
  float* st   = w + off; off += 2048;                      // stats block
  float* h1   = w + off; off += (size_t)BB * NN * NN * DD; // 67.1M
  float* h2   = w + off; off += (size_t)BB * NN * NN * DD; // 67.1M

  float *sum1 = st, *sq1 = st + 128, *sum2 = st + 256, *sq2 = st + 384;
  float *sum3 = st + 512, *sq3 = st + 640, *sum4 = st + 768, *sq4 = st + 896;
  float *A1 = st + 1024, *C1 = st + 1152, *A2 = st + 1280, *C2 = st + 1408;
  float *A3 = st + 1536, *C3 = st + 1664, *A4 = st + 1792, *C4 = st + 1920;

  hipMemsetAsync(st, 0, 1024 * sizeof(float), stream);  // zero stat accumulators

  const int Mseq = BB * T2N;       // 2048
  const int Mh   = BB * NN * NN;   // 524288
  const int Mz   = BB * NN;        // 1024

  // 1) seq = concat(eig, nf)
  concat_seq<<<BB * NN * DD / 256, 256, 0, stream>>>(eig, nf, seq);

  // 2) q/k/v projections
  gemm16<0, 0><<<dim3(Mseq / 64, DD / 16), 32, 0, stream>>>(
      seq, DD, Wq, DD, qb, DD, Mseq, DD, DD, bq, nullptr, 0, nullptr, nullptr, nullptr, 0);
  gemm16<0, 0><<<dim3(Mseq / 64, DD / 16), 32, 0, stream>>>(
      seq, DD, Wk, DD, kb, DD, Mseq, DD, DD, bk, nullptr, 0, nullptr, nullptr, nullptr, 0);
  gemm16<0, 0><<<dim3(Mseq / 64, DD / 16), 32, 0, stream>>>(
      seq, DD, Wv, DD, vb, DD, Mseq, DD, DD, bv, nullptr, 0, nullptr, nullptr, nullptr, 0);

  // 3) flash attention
  attn16<<<dim3(T2N / 16, BB * HH), 32, 0, stream>>>(qb, kb, vb, length, ob);

  // 4) seq2 = seq + o@Wo + bo
  gemm16<0, 0><<<dim3(Mseq / 64, DD / 16), 32, 0, stream>>>(
      ob, DD, Wo, DD, seq2, DD, Mseq, DD, DD, bo, seq, DD, nullptr, nullptr, nullptr, 0);

  // 5) new_e
  newe_k<<<(BB * NN * KK + 255) / 256, 256, 0, stream>>>(seq2, Wd, bd, ne);

  // 6) filters[b,k] = (U * diag(ne[b,k])) @ U^T  -> layout (B,K,N,N)
  for (int b = 0; b < BB; ++b)
    for (int kk = 0; kk < KK; ++kk) {
      const float* Ub = u + (size_t)b * NN * NN;
      gemm16<0, 1><<<dim3(NN / 64, NN / 16), 32, 0, stream>>>(
          Ub, NN, Ub, NN, filt + ((size_t)b * KK + kk) * NN * NN, NN,
          NN, NN, NN, nullptr, nullptr, 0, nullptr, nullptr,
          ne + ((size_t)b * KK + kk) * NN, 0);
    }

  // 7) h1 = bases @ W1 + b1  (ident column fused)
  h1_k<<<Mh / 4, DD, 0, stream>>>(filt, W1, b1, h1);

  // 8) BN1 stats -> affine
  colstats<<<Mh / 256, 256, 0, stream>>>(h1, Mh, 256, sum1, sq1);
  bn_fin<<<1, DD, 0, stream>>>(sum1, sq1, g1, be1, 1.0f / Mh, A1, C1);

  // 9) h2 = gelu(bn1(h1)) @ W2 + b2   (BN+gelu fused into A-load)
  gemm16<1, 0><<<dim3(Mh / 64, DD / 16), 32, 0, stream>>>(
      h1, DD, W2, DD, h2, DD, Mh, DD, DD, b2, nullptr, 0, A1, C1, nullptr, 0);

  // 10) BN2 stats -> affine
  colstats<<<Mh / 256, 256, 0, stream>>>(h2, Mh, 256, sum2, sq2);
  bn_fin<<<1, DD, 0, stream>>>(sum2, sq2, g2, be2, 1.0f / Mh, A2, C2);

  // 11) pre = gelu(nf2 @ Wp + bp)   (per-batch: nf2 rows are seq2[b, N..2N))
  for (int b = 0; b < BB; ++b)
    gemm16<0, 0><<<dim3(NN / 64, DD / 16), 32, 0, stream>>>(
        seq2 + ((size_t)b * T2N + NN) * DD, DD, Wp, DD,
        pre + (size_t)b * NN * DD, DD, NN, DD, DD, bp,
        nullptr, 0, nullptr, nullptr, nullptr, /*gelu*/ 1);

  // 12) fused softmax(j) + y einsum + residual -> nf3
  smax_y<<<BB * NN, DD, 0, stream>>>(h2, A2, C2, pre, seq2, nf3);

  // 13) final MLP: t1 = nf3@Wc1+bc1 ; BN ; t2 = relu(bn(t1))@Wc2+bc2 ; BN
  gemm16<0, 0><<<dim3(Mz / 64, DD / 16), 32, 0, stream>>>(
      nf3, DD, Wc1, DD, t1, DD, Mz, DD, DD, bc1, nullptr, 0, nullptr, nullptr, nullptr, 0);
  colstats<<<Mz / 256, 256, 0, stream>>>(t1, Mz, 256, sum3, sq3);
  bn_fin<<<1, DD, 0, stream>>>(sum3, sq3, gc1, bec1, 1.0f / Mz, A3, C3);
  gemm16<2, 0><<<dim3(Mz / 64, DD / 16), 32, 0, stream>>>(
      t1, DD, Wc2, DD, t2, DD, Mz, DD, DD, bc2, nullptr, 0, A3, C3, nullptr, 0);
  colstats<<<Mz / 256, 256, 0, stream>>>(t2, Mz, 256, sum4, sq4);
  bn_fin<<<1, DD, 0, stream>>>(sum4, sq4, gc2, bec2, 1.0f / Mz, A4, C4);

  // 14) outputs: nf_out then eig2
  fin_out<<<BB * NN * DD / 256, 256, 0, stream>>>(nf3, t2, A4, C4, seq2, out);
}